// YoloLoss_31147102830917
// MI455X (gfx1250) — compile-verified
//
#include <hip/hip_runtime.h>
#include <stdint.h>

// Problem constants (S=7, B=2, C=20, NE=30, BATCH=32768)
#define NCELL   1605632              // 32768 * 49
#define NBYTESA 192675840u           // NCELL * 30 floats * 4B = bytes per input array
#define NB      1024                 // blocks for partial kernel
#define BLOCK   256                  // 8 waves of 32
#define NWAVE   8
#define SEGB    3840u                // bytes per wave per array (32 cells * 120B)
#define SEGF    960                  // floats per wave per array

// ---- CDNA5 async global->LDS copy (ASYNCcnt path), via inline asm ----------
// Non-temporal hint: 385MB streamed once > 192MB L2 -> don't cache it anywhere.
__device__ __forceinline__ void async_ld_b64(uint32_t lds_off, const void* sbase, uint32_t voff) {
    // GVS mode: mem = SGPR64 base + 32-bit per-lane byte offset; VDST = LDS byte addr
    asm volatile("global_load_async_to_lds_b64 %0, %1, %2 offset:0 th:TH_LOAD_NT"
                 :: "v"(lds_off), "v"(voff), "s"(sbase)
                 : "memory");
}
__device__ __forceinline__ void wait_asynccnt0() {
    asm volatile("s_wait_asynccnt 0" ::: "memory");
}
__device__ __forceinline__ void wait_dscnt0() {
    asm volatile("s_wait_dscnt 0" ::: "memory");
}

__global__ __launch_bounds__(BLOCK)
void yolo_partial(const float* __restrict__ pred,
                  const float* __restrict__ targ,
                  float* __restrict__ ws) {
    // [0, 7680): pred segments (8 waves x 960 floats); [7680, 15360): targ segments
    __shared__ float sh[2 * NWAVE * SEGF];

    const int tid  = threadIdx.x;
    const int wave = tid >> 5;
    const int lane = tid & 31;

    // Low 32 bits of a generic LDS pointer == LDS byte offset (aperture in high bits).
    const uint32_t lds0  = (uint32_t)(uintptr_t)&sh[0];
    const uint32_t segP  = lds0 + (uint32_t)wave * SEGB;
    const uint32_t segT  = lds0 + (uint32_t)(NWAVE) * SEGB + (uint32_t)wave * SEGB;
    const uint32_t lane8 = (uint32_t)lane * 8u;

    const float* Pl = &sh[wave * SEGF + lane * 30];
    const float* Tl = &sh[NWAVE * SEGF + wave * SEGF + lane * 30];

    float accCls = 0.f, accBce = 0.f, accLoc = 0.f, accResp = 0.f, accCoord = 0.f;

    for (int base = blockIdx.x * BLOCK; base < NCELL; base += NB * BLOCK) {
        const int cell0 = base + wave * 32;           // wave's first cell this step
        const uint32_t src0 = (uint32_t)cell0 * 120u; // byte offset of wave region

        // Ensure prior LDS reads retired before async engine overwrites the tile.
        wait_dscnt0();

        // Stream 32 cells of prediction (3840B) and target (3840B) into LDS.
#pragma unroll
        for (int k = 0; k < 15; ++k) {
            const uint32_t rel = (uint32_t)k * 256u + lane8;
            uint32_t sp = src0 + rel;
            if (sp > NBYTESA - 8u) sp = NBYTESA - 8u;   // tail clamp (data masked later)
            async_ld_b64(segP + rel, pred, sp);
        }
#pragma unroll
        for (int k = 0; k < 15; ++k) {
            const uint32_t rel = (uint32_t)k * 256u + lane8;
            uint32_t sp = src0 + rel;
            if (sp > NBYTESA - 8u) sp = NBYTESA - 8u;
            async_ld_b64(segT + rel, targ, sp);
        }
        wait_asynccnt0();   // wave-local produce/consume: no workgroup barrier needed

        const int cell = cell0 + lane;
        if (cell < NCELL) {
            float pf[2][4], tf[2][4], confp[2], conft[2], sq[2];
#pragma unroll
            for (int b = 0; b < 2; ++b) {
                const int o = 5 * b;
                const float px = Pl[o+0], py = Pl[o+1], pw = Pl[o+2], ph = Pl[o+3];
                const float tx = Tl[o+0], ty = Tl[o+1], tw = Tl[o+2], th = Tl[o+3];
                pf[b][0] = px - pw*0.5f; pf[b][1] = py - ph*0.5f;
                pf[b][2] = px + pw*0.5f; pf[b][3] = py + ph*0.5f;
                tf[b][0] = tx - tw*0.5f; tf[b][1] = ty - th*0.5f;
                tf[b][2] = tx + tw*0.5f; tf[b][3] = ty + th*0.5f;
                confp[b] = Pl[o+4]; conft[b] = Tl[o+4];
                const float dx = px-tx, dy = py-ty, dw = pw-tw, dh = ph-th;
                sq[b] = dx*dx + dy*dy + dw*dw + dh*dh;
            }
            float iou[2][2];
#pragma unroll
            for (int i = 0; i < 2; ++i)
#pragma unroll
                for (int j = 0; j < 2; ++j) {
                    const float lox = fmaxf(pf[i][0], tf[j][0]);
                    const float loy = fmaxf(pf[i][1], tf[j][1]);
                    const float hix = fminf(pf[i][2], tf[j][2]);
                    const float hiy = fminf(pf[i][3], tf[j][3]);
                    const float inter = fmaxf(hix - lox, 0.f) * fmaxf(hiy - loy, 0.f);
                    const float aa = (pf[i][2]-pf[i][0]) * (pf[i][3]-pf[i][1]);
                    const float ab = (tf[j][2]-tf[j][0]) * (tf[j][3]-tf[j][1]);
                    iou[i][j] = inter / (aa + ab - inter + 1e-12f);
                }
            const bool coord = Tl[4] > 0.f;
            // jnp.argmax over pred axis: first index wins ties -> strict '>'
            const int i0 = (iou[1][0] > iou[0][0]) ? 1 : 0;
            const int i1 = (iou[1][1] > iou[0][1]) ? 1 : 0;
#pragma unroll
            for (int b = 0; b < 2; ++b) {
                const bool resp = coord && ((i0 == b) || (i1 == b));
                if (resp) {
                    const float cp = confp[b];
                    // logaddexp(0, cp) - cp*ct, numerically stable
                    const float bce = fmaxf(cp, 0.f) + log1pf(expf(-fabsf(cp))) - cp * conft[b];
                    accBce  += bce;
                    accLoc  += sq[b];
                    accResp += 1.f;
                }
            }
            if (coord) {
                float lg[20];
                float mx = -3.402823466e38f;
#pragma unroll
                for (int c = 0; c < 20; ++c) { lg[c] = Pl[10 + c]; mx = fmaxf(mx, lg[c]); }
                float se = 0.f;
#pragma unroll
                for (int c = 0; c < 20; ++c) se += expf(lg[c] - mx);
                const float lse = mx + logf(se);
                int cls = (int)floorf(Tl[10]);        // encode quirk: t[10] == h*S
                cls = cls < 0 ? 0 : (cls > 19 ? 19 : cls);
                accCls   += lse - lg[cls];
                accCoord += 1.f;
            }
        }
    }

    // ---- deterministic block reduction: wave shuffle, then LDS across waves ----
    float v[5] = { accCls, accBce, accLoc, accResp, accCoord };
#pragma unroll
    for (int j = 0; j < 5; ++j)
#pragma unroll
        for (int m = 16; m > 0; m >>= 1)
            v[j] += __shfl_down(v[j], m, 32);

    __syncthreads();                 // tile LDS no longer needed; reuse it
    if (lane == 0) {
#pragma unroll
        for (int j = 0; j < 5; ++j) sh[wave * 5 + j] = v[j];
    }
    __syncthreads();
    if (tid == 0) {
        float s[5] = {0.f, 0.f, 0.f, 0.f, 0.f};
        for (int w = 0; w < NWAVE; ++w)
            for (int j = 0; j < 5; ++j) s[j] += sh[w * 5 + j];
        for (int j = 0; j < 5; ++j) ws[blockIdx.x * 5 + j] = s[j];
    }
}

__global__ __launch_bounds__(256)
void yolo_final(const float* __restrict__ ws, float* __restrict__ out) {
    __shared__ float red[256];
    __shared__ float tot[5];
    const int tid = threadIdx.x;
    float s[5] = {0.f, 0.f, 0.f, 0.f, 0.f};
    for (int i = tid; i < NB; i += 256) {
#pragma unroll
        for (int j = 0; j < 5; ++j) s[j] += ws[i * 5 + j];
    }
#pragma unroll
    for (int j = 0; j < 5; ++j) {
        red[tid] = s[j];
        __syncthreads();
        for (int m = 128; m > 0; m >>= 1) {
            if (tid < m) red[tid] += red[tid + m];
            __syncthreads();
        }
        if (tid == 0) tot[j] = red[0];
        __syncthreads();
    }
    if (tid == 0) {
        const float cnt  = fmaxf(tot[3], 1.f);   // responsible-box count
        const float ccnt = fmaxf(tot[4], 1.f);   // coord-cell count
        // class_loss + contain_loss + L_COORD * loc_loss  (L_COORD = 5)
        out[0] = tot[0] / ccnt + tot[1] / cnt + 5.0f * (tot[2] / (2.0f * cnt));
    }
}

extern "C" void kernel_launch(void* const* d_in, const int* in_sizes, int n_in,
                              void* d_out, int out_size, void* d_ws, size_t ws_size,
                              hipStream_t stream) {
    const float* pred = (const float*)d_in[0];   // [32768, 49, 30] f32
    const float* targ = (const float*)d_in[1];   // [32768, 49, 30] f32
    float* out = (float*)d_out;                  // scalar f32
    float* ws  = (float*)d_ws;                   // needs NB*5*4 = 20,480 bytes

    yolo_partial<<<NB, BLOCK, 0, stream>>>(pred, targ, ws);
    yolo_final<<<1, 256, 0, stream>>>(ws, out);
}